// GraphNet_tempscale_41042707480592
// MI455X (gfx1250) — compile-verified
//
#include <hip/hip_runtime.h>
#include <hip/hip_bf16.h>
#include <stdint.h>

typedef __attribute__((ext_vector_type(2))) float v2f;
typedef __attribute__((ext_vector_type(8))) float v8f;

#define CONT   1000
#define NCAT   24
#define MAXC   100
#define NFEAT  128
#define NCLS   100
#define NNODE  1024
#define NEDGE  65536
#define BROWS  8192
#define FC_IN  8392          // NNODE*8 + 2*NCLS
#define BETA   1.1f

// ---- workspace layout (float offsets) ----
#define OFF_X     0          // 1024*128
#define OFF_H1    131072     // 1024*16
#define OFF_X1    147456     // 1024*16
#define OFF_H2    163840     // 1024*8
#define OFF_X2    172032     // 1024*8 (== flat)
#define OFF_DINV  180224     // 1024
#define OFF_CONST 181248     // 128
#define OFF_DEG   181376     // 1024   --+
#define OFF_AGG1  182400     // 1024*16  | contiguous zeroed region
#define OFF_AGG2  198784     // 1024*8 --+
#define ZERO_FLOATS (1024 + 16384 + 8192)

// ---------------- x = concat(num_x, embedding) ----------------
__global__ void k_build_x(const float* __restrict__ num_x,
                          const float* __restrict__ cat_x,
                          const float* __restrict__ emb_W,
                          const float* __restrict__ emb_b,
                          float* __restrict__ x) {
  int tid = blockIdx.x * blockDim.x + threadIdx.x;   // over 1024*128
  if (tid >= NNODE * NFEAT) return;
  int n = tid / NFEAT, f = tid % NFEAT;
  if (n < CONT) {
    x[tid] = num_x[n * NFEAT + f];
  } else {
    int k = n - CONT;
    const float* w = emb_W + ((size_t)k * NFEAT + f) * MAXC;
    const float* cx = cat_x + k * MAXC;
    float s = emb_b[k * NFEAT + f];
    #pragma unroll 4
    for (int c = 0; c < MAXC; ++c) s += cx[c] * w[c];
    x[tid] = s;
  }
}

// ---------------- degree / dinv ----------------
__global__ void k_degree(const long long* __restrict__ ei,
                         const float* __restrict__ ew,
                         float* __restrict__ deg) {
  int e = blockIdx.x * blockDim.x + threadIdx.x;
  if (e >= NEDGE) return;
  int d = (int)ei[NEDGE + e];
  atomicAdd(&deg[d], ew[e]);
}

__global__ void k_dinv(const float* __restrict__ deg, float* __restrict__ dinv) {
  int n = blockIdx.x * blockDim.x + threadIdx.x;
  if (n >= NNODE) return;
  float dt = deg[n] + 1.0f;            // self-loop weight 1
  dinv[n] = dt > 0.0f ? rsqrtf(dt) : 0.0f;
}

// ---------------- h1 = x @ W1  via f32 WMMA 16x16x4 ----------------
__global__ void k_h1_wmma(const float* __restrict__ x,
                          const float* __restrict__ W1,
                          float* __restrict__ h1) {
  int wave = threadIdx.x >> 5;
  int lane = threadIdx.x & 31;
  int mtile = blockIdx.x * 4 + wave;          // 64 tiles, 16 blocks
  int ml = lane & 15;                         // row (A) == col (B) index
  int kg = lane >> 4;                         // K-pair group
  int row = mtile * 16 + ml;
  v8f acc = {};
  #pragma unroll 8
  for (int ks = 0; ks < NFEAT / 4; ++ks) {
    int k0 = ks * 4 + kg * 2;
    v2f a, b;
    a.x = x[row * NFEAT + k0];
    a.y = x[row * NFEAT + k0 + 1];
    b.x = W1[k0 * 16 + ml];
    b.y = W1[(k0 + 1) * 16 + ml];
    acc = __builtin_amdgcn_wmma_f32_16x16x4_f32(false, a, false, b,
                                                (short)0, acc, false, false);
  }
  #pragma unroll
  for (int v = 0; v < 8; ++v) {
    int mr = mtile * 16 + v + 8 * kg;
    h1[mr * 16 + ml] = acc[v];
  }
}

// ---------------- edge aggregation (scatter-add) ----------------
template <int CH>
__global__ void k_edge_agg(const long long* __restrict__ ei,
                           const float* __restrict__ ew,
                           const float* __restrict__ dinv,
                           const float* __restrict__ h,
                           float* __restrict__ agg) {
  int e = blockIdx.x * blockDim.x + threadIdx.x;
  if (e >= NEDGE) return;
  int s = (int)ei[e];
  int d = (int)ei[NEDGE + e];
  float norm = dinv[s] * ew[e] * dinv[d];
  #pragma unroll
  for (int c = 0; c < CH; ++c)
    atomicAdd(&agg[d * CH + c], norm * h[s * CH + c]);
}

// ---------------- self-loop + bias + relu ----------------
template <int CH>
__global__ void k_finish(const float* __restrict__ agg,
                         const float* __restrict__ h,
                         const float* __restrict__ dinv,
                         const float* __restrict__ bias,
                         float* __restrict__ out) {
  int tid = blockIdx.x * blockDim.x + threadIdx.x;
  if (tid >= NNODE * CH) return;
  int n = tid / CH, c = tid % CH;
  float di = dinv[n];
  float v = agg[tid] + di * di * h[tid] + bias[c];
  out[tid] = fmaxf(v, 0.0f);
}

// ---------------- h2 = x1 @ W2 (16->8, tiny) ----------------
__global__ void k_h2(const float* __restrict__ x1,
                     const float* __restrict__ W2,
                     float* __restrict__ h2) {
  int tid = blockIdx.x * blockDim.x + threadIdx.x;
  if (tid >= NNODE * 8) return;
  int n = tid / 8, c = tid % 8;
  float s = 0.0f;
  #pragma unroll
  for (int f = 0; f < 16; ++f) s += x1[n * 16 + f] * W2[f * 8 + c];
  h2[tid] = s;
}

// ---------------- const[j] = flat.fc_W[j,100:8292] + prob.fc_W[j,8292:] + fc_b[j] ----------------
__global__ void k_const(const float* __restrict__ flat,
                        const float* __restrict__ prob,
                        const float* __restrict__ fcW,
                        const float* __restrict__ fcb,
                        float* __restrict__ constv) {
  int j = blockIdx.x;
  __shared__ float red[256];
  const float* wrow = fcW + (size_t)j * FC_IN + NCLS;
  float s = 0.0f;
  for (int i = threadIdx.x; i < FC_IN - NCLS; i += 256) {
    float v = (i < NNODE * 8) ? flat[i] : prob[i - NNODE * 8];
    s += v * wrow[i];
  }
  red[threadIdx.x] = s;
  __syncthreads();
  for (int off = 128; off > 0; off >>= 1) {
    if ((int)threadIdx.x < off) red[threadIdx.x] += red[threadIdx.x + off];
    __syncthreads();
  }
  if (threadIdx.x == 0) constv[j] = red[0] + fcb[j];
}

// ---------------- main GEMM 8192x100x100 via f32 WMMA + softplus ----------------
__global__ void k_fc_wmma(const float* __restrict__ van,
                          const float* __restrict__ fcW,
                          const float* __restrict__ constv,
                          float* __restrict__ out) {
  int wave = threadIdx.x >> 5;
  int lane = threadIdx.x & 31;
  int tile = blockIdx.x * 4 + wave;           // 3584 tiles = 512 Mtiles x 7 Ntiles
  int mtile = tile / 7;
  int ntile = tile % 7;
  int ml = lane & 15;
  int kg = lane >> 4;
  int row  = mtile * 16 + ml;                 // A row (always in range, 8192 % 16 == 0)
  int ncol = ntile * 16 + ml;                 // B column / class index
  bool nok = ncol < NCLS;
  v8f acc = {};
  #pragma unroll 5
  for (int ks = 0; ks < NCLS / 4; ++ks) {     // K = 100 = 25 * 4
    int k0 = ks * 4 + kg * 2;
    v2f a, b;
    a.x = van[row * NCLS + k0];
    a.y = van[row * NCLS + k0 + 1];
    b.x = nok ? fcW[(size_t)ncol * FC_IN + k0]     : 0.0f;
    b.y = nok ? fcW[(size_t)ncol * FC_IN + k0 + 1] : 0.0f;
    acc = __builtin_amdgcn_wmma_f32_16x16x4_f32(false, a, false, b,
                                                (short)0, acc, false, false);
  }
  if (nok) {
    float cj = constv[ncol];
    #pragma unroll
    for (int v = 0; v < 8; ++v) {
      int mr = mtile * 16 + v + 8 * kg;
      float z = (acc[v] + cj) * BETA;
      float sp = (fmaxf(z, 0.0f) + log1pf(__expf(-fabsf(z)))) / BETA;
      out[(size_t)mr * NCLS + ncol] = sp;
    }
  }
}

extern "C" void kernel_launch(void* const* d_in, const int* in_sizes, int n_in,
                              void* d_out, int out_size, void* d_ws, size_t ws_size,
                              hipStream_t stream) {
  const float*     num_x  = (const float*)d_in[0];
  const float*     cat_x  = (const float*)d_in[1];
  const float*     emb_W  = (const float*)d_in[2];
  const float*     emb_b  = (const float*)d_in[3];
  const float*     W1     = (const float*)d_in[4];
  const float*     b1     = (const float*)d_in[5];
  const float*     W2     = (const float*)d_in[6];
  const float*     b2     = (const float*)d_in[7];
  const float*     fcW    = (const float*)d_in[8];
  const float*     fcb    = (const float*)d_in[9];
  const float*     van    = (const float*)d_in[10];
  const float*     prob   = (const float*)d_in[11];
  const long long* ei     = (const long long*)d_in[12];   // int64 edge_index
  const float*     ew     = (const float*)d_in[13];
  float* out = (float*)d_out;
  float* ws  = (float*)d_ws;

  float* x     = ws + OFF_X;
  float* h1    = ws + OFF_H1;
  float* x1    = ws + OFF_X1;
  float* h2    = ws + OFF_H2;
  float* x2    = ws + OFF_X2;
  float* dinv  = ws + OFF_DINV;
  float* cst   = ws + OFF_CONST;
  float* deg   = ws + OFF_DEG;
  float* agg1  = ws + OFF_AGG1;
  float* agg2  = ws + OFF_AGG2;

  // zero deg + agg1 + agg2 (contiguous) — capture-safe
  hipMemsetAsync((void*)deg, 0, (size_t)ZERO_FLOATS * sizeof(float), stream);

  k_build_x<<<(NNODE * NFEAT + 255) / 256, 256, 0, stream>>>(num_x, cat_x, emb_W, emb_b, x);
  k_degree<<<NEDGE / 256, 256, 0, stream>>>(ei, ew, deg);
  k_dinv<<<NNODE / 256, 256, 0, stream>>>(deg, dinv);

  k_h1_wmma<<<16, 128, 0, stream>>>(x, W1, h1);               // 64 tiles, 4 waves/block
  k_edge_agg<16><<<NEDGE / 256, 256, 0, stream>>>(ei, ew, dinv, h1, agg1);
  k_finish<16><<<(NNODE * 16 + 255) / 256, 256, 0, stream>>>(agg1, h1, dinv, b1, x1);

  k_h2<<<(NNODE * 8 + 255) / 256, 256, 0, stream>>>(x1, W2, h2);
  k_edge_agg<8><<<NEDGE / 256, 256, 0, stream>>>(ei, ew, dinv, h2, agg2);
  k_finish<8><<<(NNODE * 8 + 255) / 256, 256, 0, stream>>>(agg2, h2, dinv, b2, x2);

  k_const<<<NCLS, 256, 0, stream>>>(x2, prob, fcW, fcb, cst);

  // 512 Mtiles * 7 Ntiles = 3584 tiles, 4 waves/block -> 896 blocks
  k_fc_wmma<<<896, 128, 0, stream>>>(van, fcW, cst, out);
}